// HeteroPatch_Encoding_13769665151130
// MI455X (gfx1250) — compile-verified
//
#include <hip/hip_runtime.h>
#include <hip/hip_bf16.h>
#include <math.h>

typedef __attribute__((ext_vector_type(2))) float v2f;
typedef __attribute__((ext_vector_type(8))) float v8f;

#define FEAT_DIM 128
#define TIME_DIM 100
#define IN_DIM   (FEAT_DIM + TIME_DIM)   // 228
#define OUT_DIM  128
#define NUM_TYPES 4
#define A_STRIDE 234                      // 228 padded: even (8B align), few LDS bank conflicts

// ---------------- workspace layout (ints) ----------------
// [0..3]   counts
// [4..7]   cursors
// [8..12]  off[0..4]  (64-aligned exclusive prefix of counts)
// [16.. ]  sortedIdx[sidxLen]  (-1 = padding slot)

__global__ void hp_init_kernel(int* ws_i, int sidxLen) {
    int i = blockIdx.x * blockDim.x + threadIdx.x;
    if (i < 16) ws_i[i] = 0;
    for (int j = i; j < sidxLen; j += gridDim.x * blockDim.x)
        ws_i[16 + j] = -1;
}

__global__ void hp_hist_kernel(const int* __restrict__ types, int* counts, int E) {
    __shared__ int h[NUM_TYPES];
    if (threadIdx.x < NUM_TYPES) h[threadIdx.x] = 0;
    __syncthreads();
    int i = blockIdx.x * blockDim.x + threadIdx.x;
    if (i < E) atomicAdd(&h[types[i]], 1);
    __syncthreads();
    if (threadIdx.x < NUM_TYPES) atomicAdd(&counts[threadIdx.x], h[threadIdx.x]);
}

__global__ void hp_offsets_kernel(const int* counts, int* off) {
    if (threadIdx.x == 0 && blockIdx.x == 0) {
        int o = 0;
        off[0] = 0;
        for (int t = 0; t < NUM_TYPES; t++) {
            o += ((counts[t] + 63) >> 6) << 6;   // align each segment to 64 edges
            off[t + 1] = o;
        }
    }
}

__global__ void hp_scatter_kernel(const int* __restrict__ types, const int* __restrict__ off,
                                  int* cursors, int* sidx, int E) {
    __shared__ int h[NUM_TYPES];
    __shared__ int base[NUM_TYPES];
    if (threadIdx.x < NUM_TYPES) h[threadIdx.x] = 0;
    __syncthreads();
    int i = blockIdx.x * blockDim.x + threadIdx.x;
    int ty = 0, rank = 0;
    bool valid = (i < E);
    if (valid) {
        ty = types[i];
        rank = atomicAdd(&h[ty], 1);
    }
    __syncthreads();
    if (threadIdx.x < NUM_TYPES)
        base[threadIdx.x] = atomicAdd(&cursors[threadIdx.x], h[threadIdx.x]);
    __syncthreads();
    if (valid) sidx[off[ty] + base[ty] + rank] = i;
}

// One 256-thread workgroup = 8 waves, processes 64 edges (4 M-tiles of 16) x 128 outputs.
// Wave w owns output columns [16w, 16w+16) across all 4 M-tiles.
__global__ __launch_bounds__(256)
void hp_gemm_kernel(const float* __restrict__ feats,
                    const float* __restrict__ ts,
                    const float* __restrict__ W,
                    const float* __restrict__ bias,
                    const float* __restrict__ emb,
                    const float* __restrict__ freqs,
                    const int* __restrict__ sidx,
                    const int* __restrict__ off,
                    float* __restrict__ out) {
    __shared__ float sA[64 * A_STRIDE];
    __shared__ int   sIdx[64];
    __shared__ float sTs[64];

    const int wgStart = blockIdx.x * 64;
    const int off4 = off[4];
    if (wgStart >= off4) return;                 // uniform early-exit (EXEC stays full)

    // segment type of this (single-type, 64-aligned) block
    const int ty = (wgStart >= off[1]) + (wgStart >= off[2]) + (wgStart >= off[3]);

    const int tid = threadIdx.x;

    // --- stage edge indices + timestamps ---
    if (tid < 64) {
        int idx = sidx[wgStart + tid];
        sIdx[tid] = idx;
        sTs[tid] = (idx >= 0) ? ts[idx] : 0.0f;
    }
    __syncthreads();

    // --- stage A tile: [64][228] = [feat(128) | cos(t*freq)(100)] ---
    {   // feature part: 4 threads per row, 32 floats each (float4 loads, float2 LDS stores)
        const int r = tid >> 2;
        const int q = tid & 3;
        const int idx = sIdx[r];
        float* dst = &sA[r * A_STRIDE + q * 32];
        float2* d2 = reinterpret_cast<float2*>(dst);
        if (idx >= 0) {
            const float4* s4 = reinterpret_cast<const float4*>(feats + (size_t)idx * FEAT_DIM + q * 32);
#pragma unroll
            for (int i = 0; i < 8; i++) {
                float4 v = s4[i];
                d2[2 * i + 0] = make_float2(v.x, v.y);
                d2[2 * i + 1] = make_float2(v.z, v.w);
            }
        } else {
#pragma unroll
            for (int i = 0; i < 16; i++) d2[i] = make_float2(0.0f, 0.0f);
        }
    }
    {   // time-encoding part
        const float* fq = freqs + ty * TIME_DIM;
        for (int i = tid; i < 64 * TIME_DIM; i += 256) {
            int r = i / TIME_DIM;
            int c = i - r * TIME_DIM;
            float v = (sIdx[r] >= 0) ? cosf(sTs[r] * fq[c]) : 0.0f;
            sA[r * A_STRIDE + FEAT_DIM + c] = v;
        }
    }
    __syncthreads();

    const int lane = tid & 31;
    const int wave = tid >> 5;
    const int colBase = wave * 16;       // this wave's 16 output columns
    const int n = lane & 15;
    const int half = lane >> 4;          // 0: K even pair base, 1: K+2 pair base

    // accumulator init with per-type bias + type embedding (same for every row m)
    const float cinit = bias[ty * OUT_DIM + colBase + n] + emb[ty * OUT_DIM + colBase + n];
    v8f acc[4];
#pragma unroll
    for (int t = 0; t < 4; t++)
#pragma unroll
        for (int v = 0; v < 8; v++) acc[t][v] = cinit;

    // B base for this lane's output column
    const float* Wt = W + (size_t)ty * IN_DIM * OUT_DIM + colBase + n;
    const int mrow = lane & 15;

    for (int k0 = 0; k0 < IN_DIM; k0 += 4) {
        const int bk = k0 + 2 * half;
        v2f bv;
        bv[0] = Wt[(size_t)bk * OUT_DIM];
        bv[1] = Wt[(size_t)(bk + 1) * OUT_DIM];
#pragma unroll
        for (int t = 0; t < 4; t++) {
            const float2 af = *reinterpret_cast<const float2*>(
                &sA[(t * 16 + mrow) * A_STRIDE + k0 + 2 * half]);   // ds_load_b64
            v2f av; av[0] = af.x; av[1] = af.y;
            acc[t] = __builtin_amdgcn_wmma_f32_16x16x4_f32(
                false, av, false, bv, (short)0, acc[t], false, false);
        }
    }

    // scatter results back by original edge index; padding rows (idx<0) are dropped
#pragma unroll
    for (int t = 0; t < 4; t++) {
#pragma unroll
        for (int v = 0; v < 8; v++) {
            const int m = v + 8 * half;
            const int idx = sIdx[t * 16 + m];
            if (idx >= 0) out[(size_t)idx * OUT_DIM + colBase + n] = acc[t][v];
        }
    }
}

extern "C" void kernel_launch(void* const* d_in, const int* in_sizes, int n_in,
                              void* d_out, int out_size, void* d_ws, size_t ws_size,
                              hipStream_t stream) {
    const float* feats = (const float*)d_in[0];
    const float* ets   = (const float*)d_in[1];
    const int*   types = (const int*)d_in[2];
    const float* W     = (const float*)d_in[3];
    const float* b     = (const float*)d_in[4];
    const float* emb   = (const float*)d_in[5];
    const float* freqs = (const float*)d_in[6];
    float* out = (float*)d_out;

    const int E = in_sizes[1];                       // edge_ts count
    // worst-case padded length: E + 4*63, rounded up to 64
    const int sidxLen = (((E + NUM_TYPES * 63) + 63) / 64) * 64;

    int* ws_i    = (int*)d_ws;
    int* counts  = ws_i;
    int* cursors = ws_i + 4;
    int* off     = ws_i + 8;
    int* sidx    = ws_i + 16;

    const int thr = 256;
    hp_init_kernel<<<(sidxLen + thr - 1) / thr, thr, 0, stream>>>(ws_i, sidxLen);
    hp_hist_kernel<<<(E + thr - 1) / thr, thr, 0, stream>>>(types, counts, E);
    hp_offsets_kernel<<<1, 64, 0, stream>>>(counts, off);
    hp_scatter_kernel<<<(E + thr - 1) / thr, thr, 0, stream>>>(types, off, cursors, sidx, E);
    hp_gemm_kernel<<<sidxLen / 64, thr, 0, stream>>>(feats, ets, W, b, emb, freqs, sidx, off, out);
}